// Model_17557826306793
// MI455X (gfx1250) — compile-verified
//
#include <hip/hip_runtime.h>
#include <hip/hip_bf16.h>
#include <math.h>

#define TB      16
#define LSEQ    2048
#define ENCIN   21
#define HDIM    256
#define NST     32
#define NLAYERS 4
#define PREDLEN 720
#define COUT    21
#define CHUNK   64
#define NCHUNK  (LSEQ / CHUNK)   // 32

#define LDS_STRIDE (HDIM + 4)    // 260 floats: 16B-aligned rows, rows 4 banks apart

typedef __attribute__((ext_vector_type(2))) float v2f;
typedef __attribute__((ext_vector_type(4))) float v4f;
typedef __attribute__((ext_vector_type(8))) float v8f;

// ---------------- encoder: h = x @ enc_w.T + enc_b  (K=21, trivial) ----------------
__global__ __launch_bounds__(256) void encoder_kernel(const float* __restrict__ x,
    const float* __restrict__ w, const float* __restrict__ b, float* __restrict__ out)
{
    int id  = blockIdx.x * 256 + threadIdx.x;   // row*HDIM + hc
    int row = id >> 8;
    int hc  = id & 255;
    const float* xr = x + (size_t)row * ENCIN;
    const float* wr = w + (size_t)hc * ENCIN;
    float acc = b[hc];
#pragma unroll
    for (int e = 0; e < ENCIN; e++) acc = fmaf(xr[e], wr[e], acc);
    out[id] = acc;
}

// ---------------- per-layer SSM parameter prep: lambda, Cd (ZOH), lambda^CHUNK ----------------
__global__ __launch_bounds__(256) void s4_prep_kernel(const float* __restrict__ log_dt,
    const float* __restrict__ A_re, const float* __restrict__ A_im,
    const float* __restrict__ C_re, const float* __restrict__ C_im,
    float* __restrict__ lam, float* __restrict__ cd, float* __restrict__ lamT, int layer)
{
    int idx = blockIdx.x * 256 + threadIdx.x;   // h*NST + n
    if (idx >= HDIM * NST) return;
    int h   = idx / NST;
    int off = layer * HDIM * NST + idx;
    float dt  = expf(log_dt[layer * HDIM + h]);
    float ar  = -expf(A_re[off]);
    float ai  = A_im[off];
    float dre = ar * dt, dim = ai * dt;
    float er  = expf(dre);
    float lr  = er * cosf(dim), li = er * sinf(dim);
    // Cd = (Cr + i Ci) * (lam - 1) / A
    float den = ar * ar + ai * ai;
    float nr  = lr - 1.f, ni = li;
    float qr  = (nr * ar + ni * ai) / den;
    float qi  = (ni * ar - nr * ai) / den;
    float crv = C_re[off], civ = C_im[off];
    lam[2 * idx]     = lr;
    lam[2 * idx + 1] = li;
    cd[2 * idx]      = crv * qr - civ * qi;
    cd[2 * idx + 1]  = crv * qi + civ * qr;
    float erT = expf(dre * (float)CHUNK);
    lamT[2 * idx]     = erT * cosf(dim * (float)CHUNK);
    lamT[2 * idx + 1] = erT * sinf(dim * (float)CHUNK);
}

// ---------------- pass 1: per-chunk end-state from zero state ----------------
__global__ __launch_bounds__(256) void s4_chunkstate_kernel(const float* __restrict__ xin,
    const float* __restrict__ lam, float* __restrict__ u)
{
    int bc = blockIdx.x;                 // b*NCHUNK + c
    int h  = threadIdx.x;
    int b  = bc / NCHUNK, c = bc % NCHUNK;
    float lr[NST], li[NST], sr[NST], si[NST];
#pragma unroll
    for (int n = 0; n < NST; n++) {
        lr[n] = lam[(h * NST + n) * 2];
        li[n] = lam[(h * NST + n) * 2 + 1];
        sr[n] = 0.f; si[n] = 0.f;
    }
    const float* xp = xin + ((size_t)b * LSEQ + (size_t)c * CHUNK) * HDIM + h;
#pragma unroll 1
    for (int t = 0; t < CHUNK; t++) {
        float xv = xp[(size_t)t * HDIM];
#pragma unroll
        for (int n = 0; n < NST; n++) {
            float nsr = fmaf(lr[n], sr[n], fmaf(-li[n], si[n], xv));
            float nsi = fmaf(lr[n], si[n], li[n] * sr[n]);
            sr[n] = nsr; si[n] = nsi;
        }
    }
    float* up = u + ((size_t)bc * HDIM + h) * NST * 2;
#pragma unroll
    for (int n = 0; n < NST; n++) { up[2 * n] = sr[n]; up[2 * n + 1] = si[n]; }
}

// ---------------- pass 2: sequential inter-chunk scan (in-place: u -> chunk carry-in states) ----
__global__ __launch_bounds__(256) void s4_scan_kernel(const float* __restrict__ lamT,
    float* __restrict__ u)
{
    int idx = blockIdx.x * 256 + threadIdx.x;   // b*(H*NST) + hn
    int b   = idx / (HDIM * NST);
    int hn  = idx % (HDIM * NST);
    float Lr = lamT[2 * hn], Li = lamT[2 * hn + 1];
    float sr = 0.f, si = 0.f;
#pragma unroll 1
    for (int c = 0; c < NCHUNK; c++) {
        float* p = u + (((size_t)(b * NCHUNK + c) * HDIM * NST) + hn) * 2;
        float ur = p[0], ui = p[1];
        p[0] = sr; p[1] = si;                        // carry-in state for chunk c
        float nsr = fmaf(Lr, sr, fmaf(-Li, si, ur)); // s = lamT*s + u
        float nsi = fmaf(Lr, si, fmaf( Li, sr, ui));
        sr = nsr; si = nsi;
    }
}

// ---------------- pass 3: replay chunk from carry-in, emit y = gelu(2Re(Cd.s) + D*x) ----------
__global__ __launch_bounds__(256) void s4_output_kernel(const float* __restrict__ xin,
    const float* __restrict__ lam, const float* __restrict__ cd,
    const float* __restrict__ u, const float* __restrict__ Dvec,
    float* __restrict__ yout)
{
    int bc = blockIdx.x;
    int h  = threadIdx.x;
    int b  = bc / NCHUNK, c = bc % NCHUNK;
    float lr[NST], li[NST], cr[NST], ci[NST], sr[NST], si[NST];
    const float* up = u + ((size_t)bc * HDIM + h) * NST * 2;
#pragma unroll
    for (int n = 0; n < NST; n++) {
        lr[n] = lam[(h * NST + n) * 2];
        li[n] = lam[(h * NST + n) * 2 + 1];
        cr[n] = cd[(h * NST + n) * 2];
        ci[n] = cd[(h * NST + n) * 2 + 1];
        sr[n] = up[2 * n];
        si[n] = up[2 * n + 1];
    }
    float dskip = Dvec[h];
    const float* xp = xin  + ((size_t)b * LSEQ + (size_t)c * CHUNK) * HDIM + h;
    float*       yp = yout + ((size_t)b * LSEQ + (size_t)c * CHUNK) * HDIM + h;
#pragma unroll 1
    for (int t = 0; t < CHUNK; t++) {
        float xv  = xp[(size_t)t * HDIM];
        float acc = 0.f;
#pragma unroll
        for (int n = 0; n < NST; n++) {
            float nsr = fmaf(lr[n], sr[n], fmaf(-li[n], si[n], xv));
            float nsi = fmaf(lr[n], si[n], li[n] * sr[n]);
            sr[n] = nsr; si[n] = nsi;
            acc = fmaf(cr[n], nsr, fmaf(-ci[n], nsi, acc));
        }
        float v = fmaf(2.f, acc, dskip * xv);
        // exact GeLU: 0.5*v*(1+erf(v/sqrt(2)))
        yp[(size_t)t * HDIM] = 0.5f * v * (1.f + erff(v * 0.70710678118654752f));
    }
}

// ---------------- fused GLU projection: Z = Y @ Wout.T + bout; out = Z[:,:H]*sigmoid(Z[:,H:]) --
// fp32 WMMA 16x16x4. Block = 32 rows x 512 cols. Y tile staged in LDS once via the CDNA5
// async global->LDS path (no VGPR staging, tracked by ASYNCcnt); each wave: 2 m-tiles x
// 2 col bands x {lo,hi} = 8 accumulators -> 8 WMMAs per 6 operand loads.
__global__ __launch_bounds__(256) void glu_wmma_kernel(
    const float* __restrict__ Y, const float* __restrict__ Wout,
    const float* __restrict__ bout, float* __restrict__ Hout)
{
    __shared__ float lds[32 * LDS_STRIDE];

    const int lane = threadIdx.x;      // 0..31
    const int wave = threadIdx.y;      // 0..7
    const int tid  = wave * 32 + lane;
    const int row0 = blockIdx.x * 32;  // 32-row super-tile

    // ---- async stage of 32x256 Y tile into LDS: 128B contiguous per thread, 8x B128 ----
    {
        const int r  = tid >> 3;        // 0..31
        const int c0 = (tid & 7) * 32;  // 0..224
        const float* src = Y + (size_t)(row0 + r) * HDIM + c0;
        unsigned int dst = (unsigned int)(uintptr_t)(void*)(lds + r * LDS_STRIDE + c0);
#pragma unroll
        for (int i = 0; i < 8; i++) {
            asm volatile("global_load_async_to_lds_b128 %0, %1, off"
                         :: "v"(dst + i * 16), "v"(src + i * 4)
                         : "memory");
        }
        asm volatile("s_wait_asynccnt 0x0" ::: "memory");
    }
    __syncthreads();

    const int l15   = lane & 15;
    const int khalf = lane >> 4;       // lanes 16..31 hold K=2,3
    const int nA    = wave * 16 + l15; // first-half column, band 0..127
    const int nB    = nA + 128;        // first-half column, band 128..255

    v8f aA0 = {}, aA1 = {}, aB0 = {}, aB1 = {};  // m-tile 0
    v8f bA0 = {}, bA1 = {}, bB0 = {}, bB1 = {};  // m-tile 1
    {
        float x0 = bout[nA], x1 = bout[nA + HDIM];
        float x2 = bout[nB], x3 = bout[nB + HDIM];
#pragma unroll
        for (int i = 0; i < 8; i++) {
            aA0[i] = x0; aA1[i] = x1; aB0[i] = x2; aB1[i] = x3;
            bA0[i] = x0; bA1[i] = x1; bB0[i] = x2; bB1[i] = x3;
        }
    }

    const float* wA0 = Wout + (size_t)nA * HDIM;
    const float* wA1 = Wout + (size_t)(nA + HDIM) * HDIM;
    const float* wB0 = Wout + (size_t)nB * HDIM;
    const float* wB1 = Wout + (size_t)(nB + HDIM) * HDIM;
    const float* a0p = lds + l15 * LDS_STRIDE;          // m-tile 0 rows
    const float* a1p = lds + (16 + l15) * LDS_STRIDE;   // m-tile 1 rows

#pragma unroll 2
    for (int k0 = 0; k0 < HDIM; k0 += 4) {
        const int ka = k0 + 2 * khalf;
        v2f a0 = *(const v2f*)(a0p + ka);     // LDS A fragments (ds_load_2addr_b64)
        v2f a1 = *(const v2f*)(a1p + ka);
        v2f w0 = *(const v2f*)(wA0 + ka);     // B(k,n)=Wout[n,k]: contiguous in k, L2-resident
        v2f w1 = *(const v2f*)(wA1 + ka);
        v2f w2 = *(const v2f*)(wB0 + ka);
        v2f w3 = *(const v2f*)(wB1 + ka);
        aA0 = __builtin_amdgcn_wmma_f32_16x16x4_f32(false, a0, false, w0, (short)0, aA0, false, false);
        bA0 = __builtin_amdgcn_wmma_f32_16x16x4_f32(false, a1, false, w0, (short)0, bA0, false, false);
        aA1 = __builtin_amdgcn_wmma_f32_16x16x4_f32(false, a0, false, w1, (short)0, aA1, false, false);
        bA1 = __builtin_amdgcn_wmma_f32_16x16x4_f32(false, a1, false, w1, (short)0, bA1, false, false);
        aB0 = __builtin_amdgcn_wmma_f32_16x16x4_f32(false, a0, false, w2, (short)0, aB0, false, false);
        bB0 = __builtin_amdgcn_wmma_f32_16x16x4_f32(false, a1, false, w2, (short)0, bB0, false, false);
        aB1 = __builtin_amdgcn_wmma_f32_16x16x4_f32(false, a0, false, w3, (short)0, aB1, false, false);
        bB1 = __builtin_amdgcn_wmma_f32_16x16x4_f32(false, a1, false, w3, (short)0, bB1, false, false);
    }

    // C/D layout: lanes 0-15 -> M = vgpr; lanes 16-31 -> M = vgpr+8; N = lane%16
    const int mr0 = row0 + (khalf ? 8 : 0);        // m-tile 0 rows
    const int mr1 = mr0 + 16;                      // m-tile 1 rows
#pragma unroll
    for (int i = 0; i < 8; i++) {
        size_t m0 = (size_t)(mr0 + i), m1 = (size_t)(mr1 + i);
        Hout[m0 * HDIM + nA] = aA0[i] * (1.f / (1.f + expf(-aA1[i])));
        Hout[m0 * HDIM + nB] = aB0[i] * (1.f / (1.f + expf(-aB1[i])));
        Hout[m1 * HDIM + nA] = bA0[i] * (1.f / (1.f + expf(-bA1[i])));
        Hout[m1 * HDIM + nB] = bB0[i] * (1.f / (1.f + expf(-bB1[i])));
    }
}

// ---------------- decoder: out = h[:, -720:, :] @ dec_w.T + dec_b ----------------
__global__ __launch_bounds__(256) void decoder_kernel(const float* __restrict__ hin,
    const float* __restrict__ w, const float* __restrict__ b, float* __restrict__ out)
{
    int id = blockIdx.x * 256 + threadIdx.x;
    if (id >= TB * PREDLEN * COUT) return;
    int cc = id % COUT;
    int rp = id / COUT;                // b*PREDLEN + p
    int bb = rp / PREDLEN, p = rp % PREDLEN;
    const float* hr = hin + ((size_t)bb * LSEQ + (LSEQ - PREDLEN) + p) * HDIM;
    const float* wr = w + (size_t)cc * HDIM;
    float acc = b[cc];
#pragma unroll 4
    for (int k = 0; k < HDIM; k++) acc = fmaf(hr[k], wr[k], acc);
    out[id] = acc;
}

extern "C" void kernel_launch(void* const* d_in, const int* in_sizes, int n_in,
                              void* d_out, int out_size, void* d_ws, size_t ws_size,
                              hipStream_t stream)
{
    const float* x      = (const float*)d_in[0];
    const float* enc_w  = (const float*)d_in[1];
    const float* enc_b  = (const float*)d_in[2];
    const float* log_dt = (const float*)d_in[3];
    const float* A_re   = (const float*)d_in[4];
    const float* A_im   = (const float*)d_in[5];
    const float* C_re   = (const float*)d_in[6];
    const float* C_im   = (const float*)d_in[7];
    const float* Dp     = (const float*)d_in[8];
    const float* Wout   = (const float*)d_in[9];
    const float* bout   = (const float*)d_in[10];
    const float* dec_w  = (const float*)d_in[11];
    const float* dec_b  = (const float*)d_in[12];
    float* out = (float*)d_out;

    float* ws = (float*)d_ws;
    const size_t SZH = (size_t)TB * LSEQ * HDIM;                 // 8.39M floats
    const size_t SZU = (size_t)TB * NCHUNK * HDIM * NST * 2;     // 8.39M floats
    float* hA    = ws;
    float* hB    = hA + SZH;
    float* ybuf  = hB + SZH;
    float* ubuf  = ybuf + SZH;
    float* plam  = ubuf + SZU;
    float* pcd   = plam + (size_t)HDIM * NST * 2;
    float* plamT = pcd  + (size_t)HDIM * NST * 2;

    encoder_kernel<<<(TB * LSEQ * HDIM) / 256, 256, 0, stream>>>(x, enc_w, enc_b, hA);

    float* cur = hA;
    float* nxt = hB;
    for (int l = 0; l < NLAYERS; l++) {
        s4_prep_kernel<<<(HDIM * NST + 255) / 256, 256, 0, stream>>>(
            log_dt, A_re, A_im, C_re, C_im, plam, pcd, plamT, l);
        s4_chunkstate_kernel<<<TB * NCHUNK, 256, 0, stream>>>(cur, plam, ubuf);
        s4_scan_kernel<<<(TB * HDIM * NST) / 256, 256, 0, stream>>>(plamT, ubuf);
        s4_output_kernel<<<TB * NCHUNK, 256, 0, stream>>>(
            cur, plam, pcd, ubuf, Dp + (size_t)l * HDIM, ybuf);
        glu_wmma_kernel<<<(TB * LSEQ) / 32, dim3(32, 8), 0, stream>>>(
            ybuf, Wout + (size_t)l * 2 * HDIM * HDIM, bout + (size_t)l * 2 * HDIM, nxt);
        float* t = cur; cur = nxt; nxt = t;
    }

    decoder_kernel<<<(TB * PREDLEN * COUT + 255) / 256, 256, 0, stream>>>(cur, dec_w, dec_b, out);
}